// BasisDense_31868657336526
// MI455X (gfx1250) — compile-verified
//
#include <hip/hip_runtime.h>

// y[b,u] = sum_{d,k} x[b,d]*kernel[d,u,k]*c_prob[b,k] + bias[u]
// GEMM over composed index c = d*8+k (Kgemm = 16384):
//   A[b,c] = x[b,c>>3]*c_prob[b,c&7]   (built on the fly in VALU)
//   W[c,u] = kernel[c>>3, u, c&7]      (double-buffered LDS via async-to-LDS)
// Full f32 precision via V_WMMA_F32_16X16X4_F32 (problem is bandwidth-bound;
// x+kernel = 160 MB fits the 192 MB L2, so HBM sees one streaming pass).

typedef __attribute__((ext_vector_type(2))) float v2f;
typedef __attribute__((ext_vector_type(4))) float v4f;
typedef __attribute__((ext_vector_type(8))) float v8f;
typedef int gv4i __attribute__((vector_size(16)));   // matches builtin prototype

#define B_SZ 4096
#define D_SZ 2048
#define U_SZ 2048
#define K_SZ 8
#define KCOMP (D_SZ * K_SZ)      // 16384
#define KC 64                    // composed-k chunk = 8 d * 8 k
#define NCHUNK (KCOMP / KC)      // 256

#define BM 128
#define BN 128
#define CHUNK_FLOATS (KC * BN)   // 8192 floats = 32 KB per buffer

#ifndef __has_builtin
#define __has_builtin(x) 0
#endif

#if __has_builtin(__builtin_amdgcn_global_load_async_to_lds_b128)
#define ASYNC_LDS 1
#else
#define ASYNC_LDS 0
#endif

__device__ __forceinline__ void copy16(const float* src, float* dst) {
#if ASYNC_LDS
    // GLOBAL_LOAD_ASYNC_TO_LDS_B128: per-lane 16B global -> LDS, ASYNCcnt-tracked
    __builtin_amdgcn_global_load_async_to_lds_b128(
        (__attribute__((address_space(1))) gv4i*)src,
        (__attribute__((address_space(3))) gv4i*)dst,
        0, 0);
#else
    *(v4f*)dst = *(const v4f*)src;
#endif
}

__device__ __forceinline__ void wait_async_done() {
#if ASYNC_LDS
#if __has_builtin(__builtin_amdgcn_s_wait_asynccnt)
    __builtin_amdgcn_s_wait_asynccnt(0);
#else
    asm volatile("s_wait_asynccnt 0x0" ::: "memory");
#endif
#endif
}

// Stage one W chunk (8 d-segments of 1024 contiguous floats) into LDS.
// Thread tid copies floats [tid*4, tid*4+3] of each segment: per-wave 512B
// contiguous on both sides, 16B aligned.
__device__ __forceinline__ void issue_copy(const float* __restrict__ kern,
                                           int block_u, int d0,
                                           float* dstbase, int tid) {
    #pragma unroll
    for (int j = 0; j < 8; ++j) {
        const float* src = kern + (size_t)(d0 + j) * (U_SZ * K_SZ)
                                + (size_t)block_u * K_SZ + tid * 4;
        copy16(src, dstbase + j * 1024 + tid * 4);
    }
}

// 16 WMMA steps over one KC=64 chunk; 8 accumulators (4 m-frags x 2 n-frags).
__device__ __forceinline__ void compute_chunk(const float* lds,
                                              const float* const xrow[4],
                                              const float cpsel[4][4],
                                              int d0, int wave_n, int l15, int hi,
                                              v8f acc[4][2]) {
    // x for the chunk's 8 consecutive d's, per m-frag (two b128 loads each)
    v4f xa[4], xb[4];
    #pragma unroll
    for (int f = 0; f < 4; ++f) {
        xa[f] = *(const v4f*)(xrow[f] + d0);
        xb[f] = *(const v4f*)(xrow[f] + d0 + 4);
    }
    #pragma unroll
    for (int t = 0; t < 16; ++t) {
        const int dl = t >> 1;          // local d
        const int kb = (t & 1) * 4;     // k base within d

        // B frag: lane holds K = 2*hi + v (v = vgpr), N = l15  (ds_load_b64)
        v2f bfrag[2];
        #pragma unroll
        for (int g = 0; g < 2; ++g) {
            const int ul = wave_n + g * 16 + l15;
            bfrag[g] = *(const v2f*)(&lds[dl * (BN * K_SZ) + ul * K_SZ + kb + 2 * hi]);
        }
        #pragma unroll
        for (int f = 0; f < 4; ++f) {
            const float xv = (dl < 4) ? xa[f][dl] : xb[f][dl - 4];
            v2f afrag;
            afrag[0] = xv * cpsel[f][(t & 1) * 2 + 0];
            afrag[1] = xv * cpsel[f][(t & 1) * 2 + 1];
            #pragma unroll
            for (int g = 0; g < 2; ++g) {
                acc[f][g] = __builtin_amdgcn_wmma_f32_16x16x4_f32(
                    false, afrag, false, bfrag[g],
                    (short)0, acc[f][g], false, false);
            }
        }
    }
}

__global__ __launch_bounds__(256)
void basis_dense_wmma(const float* __restrict__ x,
                      const float* __restrict__ c_prob,
                      const float* __restrict__ kern,
                      const float* __restrict__ bias,
                      float* __restrict__ y)
{
    __shared__ float ldsW[2][CHUNK_FLOATS];   // 64 KB double buffer

    const int tid  = threadIdx.x;
    const int lane = tid & 31;          // wave32
    const int wave = tid >> 5;          // 8 waves
    const int hi   = (lane >> 4) & 1;
    const int l15  = lane & 15;

    const int block_b = blockIdx.x * BM;
    const int block_u = blockIdx.y * BN;
    const int wave_m  = (wave & 1) * 64;
    const int wave_n  = (wave >> 1) * 32;

    v8f acc[4][2];
    #pragma unroll
    for (int f = 0; f < 4; ++f)
        #pragma unroll
        for (int g = 0; g < 2; ++g)
            acc[f][g] = v8f{};

    // Loop-invariant per-lane c_prob selections + x row pointers.
    float cpsel[4][4];
    const float* xrow[4];
    #pragma unroll
    for (int f = 0; f < 4; ++f) {
        const int row = block_b + wave_m + f * 16 + l15;
        const float* cpr = c_prob + (size_t)row * K_SZ;
        v4f c0 = *(const v4f*)(cpr);
        v4f c1 = *(const v4f*)(cpr + 4);
        cpsel[f][0] = hi ? c0[2] : c0[0];   // kb=0, v=0
        cpsel[f][1] = hi ? c0[3] : c0[1];   // kb=0, v=1
        cpsel[f][2] = hi ? c1[2] : c1[0];   // kb=4, v=0
        cpsel[f][3] = hi ? c1[3] : c1[1];   // kb=4, v=1
        xrow[f] = x + (size_t)row * D_SZ;
    }

    // Prologue: chunk 0 -> buffer 0
    issue_copy(kern, block_u, 0, &ldsW[0][0], tid);

    // Double-buffered main loop, unrolled by 2 so buffer parity is static.
    for (int pair = 0; pair < NCHUNK / 2; ++pair) {
        const int ci0 = 2 * pair;

        wait_async_done();      // our buf0 loads complete
        __syncthreads();        // all waves' buf0 loads done; buf1 reads done
        issue_copy(kern, block_u, (ci0 + 1) * 8, &ldsW[1][0], tid);
        compute_chunk(&ldsW[0][0], xrow, cpsel, ci0 * 8, wave_n, l15, hi, acc);

        wait_async_done();      // our buf1 loads complete
        __syncthreads();        // all waves' buf1 loads done; buf0 reads done
        if (ci0 + 2 < NCHUNK)
            issue_copy(kern, block_u, (ci0 + 2) * 8, &ldsW[0][0], tid);
        compute_chunk(&ldsW[1][0], xrow, cpsel, (ci0 + 1) * 8, wave_n, l15, hi, acc);
    }

    // Epilogue: C/D layout VGPR r -> M = r + 8*hi, N = l15. Bias add, NT store.
    #pragma unroll
    for (int g = 0; g < 2; ++g) {
        const int u  = block_u + wave_n + g * 16 + l15;
        const float bv = bias[u];
        #pragma unroll
        for (int f = 0; f < 4; ++f) {
            const int brow = block_b + wave_m + f * 16 + 8 * hi;
            #pragma unroll
            for (int r = 0; r < 8; ++r) {
                __builtin_nontemporal_store(acc[f][g][r] + bv,
                                            &y[(size_t)(brow + r) * U_SZ + u]);
            }
        }
    }
}

extern "C" void kernel_launch(void* const* d_in, const int* in_sizes, int n_in,
                              void* d_out, int out_size, void* d_ws, size_t ws_size,
                              hipStream_t stream) {
    const float* x      = (const float*)d_in[0];  // [4096, 2048]
    const float* c_prob = (const float*)d_in[1];  // [4096, 8]
    const float* kern   = (const float*)d_in[2];  // [2048, 2048, 8]
    const float* bias   = (const float*)d_in[3];  // [2048]
    float* y = (float*)d_out;                     // [4096, 2048]

    dim3 grid(B_SZ / BM, U_SZ / BN);  // 32 x 16
    dim3 block(256);
    basis_dense_wmma<<<grid, block, 0, stream>>>(x, c_prob, kern, bias, y);
}